// SimpleFLD_45011257262254
// MI455X (gfx1250) — compile-verified
//
#include <hip/hip_runtime.h>
#include <hip/hip_bf16.h>
#include <math.h>

#define B_DIM  32
#define T_DIM  2048
#define C_DIM  128
#define E_DIM  256
#define L_DIM  512
#define TY_DIM 128
#define PH     24
#define NCH    8
#define TCH    (T_DIM / NCH)   // 256 timesteps per chunk

typedef float v2f __attribute__((ext_vector_type(2)));
typedef float v8f __attribute__((ext_vector_type(8)));

// ---------------------------------------------------------------------------
// Kernel A: per-(chunk, batch) phase-bucketed masked sums via LDS float atomics
// grid (NCH, B), 256 threads. X/M loaded as float4/int4 (fully coalesced).
// ---------------------------------------------------------------------------
__global__ __launch_bounds__(256) void cycle_accum_kernel(
    const float* __restrict__ ts, const float* __restrict__ X,
    const int* __restrict__ M,
    float* __restrict__ part_sum, float* __restrict__ part_cnt)
{
    __shared__ float s[PH * C_DIM];
    __shared__ float cn[PH * C_DIM];
    const int k = blockIdx.x, b = blockIdx.y, tid = threadIdx.x;

    for (int i = tid; i < PH * C_DIM; i += 256) { s[i] = 0.f; cn[i] = 0.f; }
    __syncthreads();

    const int trow = tid >> 5;          // 0..7
    const int c4   = (tid & 31) * 4;    // channel group
    const int tb   = k * TCH;

    for (int i = 0; i < TCH / 8; ++i) {
        const int t = tb + trow + i * 8;
        const float tv = ts[b * T_DIM + t];                // wave-uniform broadcast
        const int p = ((int)floorf(tv)) % PH;
        const size_t base = ((size_t)(b * T_DIM + t)) * C_DIM + c4;
        const float4 xv = *(const float4*)(X + base);
        const int4   mv = *(const int4*)(M + base);
        float* sp = s  + p * C_DIM + c4;
        float* cp = cn + p * C_DIM + c4;
        float m0 = (float)mv.x, m1 = (float)mv.y, m2 = (float)mv.z, m3 = (float)mv.w;
        atomicAdd(sp + 0, xv.x * m0); atomicAdd(cp + 0, m0);
        atomicAdd(sp + 1, xv.y * m1); atomicAdd(cp + 1, m1);
        atomicAdd(sp + 2, xv.z * m2); atomicAdd(cp + 2, m2);
        atomicAdd(sp + 3, xv.w * m3); atomicAdd(cp + 3, m3);
    }
    __syncthreads();

    float* ps = part_sum + ((size_t)(k * B_DIM + b)) * PH * C_DIM;
    float* pc = part_cnt + ((size_t)(k * B_DIM + b)) * PH * C_DIM;
    for (int i = tid; i < PH * C_DIM; i += 256) { ps[i] = s[i]; pc[i] = cn[i]; }
}

// ---------------------------------------------------------------------------
// Kernel B: reduce partials over chunks, divide -> c_base [B,24,C]
// 98304 elements, 384 blocks x 256
// ---------------------------------------------------------------------------
__global__ __launch_bounds__(256) void cbase_reduce_kernel(
    const float* __restrict__ part_sum, const float* __restrict__ part_cnt,
    float* __restrict__ c_base)
{
    const int idx = blockIdx.x * 256 + threadIdx.x;   // < B*PH*C = 98304
    float s = 0.f, c = 0.f;
    for (int k = 0; k < NCH; ++k) {
        s += part_sum[(size_t)k * (B_DIM * PH * C_DIM) + idx];
        c += part_cnt[(size_t)k * (B_DIM * PH * C_DIM) + idx];
    }
    c_base[idx] = s / fmaxf(c, 1.0f);
}

// ---------------------------------------------------------------------------
// Kernel C: partial time-embedding pool: sum_t (v + sin v), v = t*w[e]+b[e]
// grid (NCH, B), 256 threads (e = tid)
// ---------------------------------------------------------------------------
__global__ __launch_bounds__(256) void time_embed_kernel(
    const float* __restrict__ ts, const float* __restrict__ tw,
    const float* __restrict__ tb, float* __restrict__ pooled_part)
{
    const int k = blockIdx.x, b = blockIdx.y, e = threadIdx.x;
    const float w = tw[e], bb = tb[e];
    float acc = 0.f;
    const int tbase = k * TCH;
    for (int i = 0; i < TCH; ++i) {
        const float t = ts[b * T_DIM + tbase + i];
        const float v = fmaf(t, w, bb);
        acc += v + sinf(v);
    }
    pooled_part[((size_t)(k * B_DIM + b)) * E_DIM + e] = acc;
}

// ---------------------------------------------------------------------------
// Kernel D1: finalize pooled (mean + channel-embed mean) into LDS (transposed,
// pitch 33 to dodge bank conflicts), then GEMM1 h = relu(pooled @ W1 + b1)
// via V_WMMA_F32_16X16X4_F32. grid 8 blocks x 256 (8 waves: 2 Mtiles x 4 Ntiles,
// each block owns a 64-wide slice of L=512). Writes h TRANSPOSED [L][B].
// ---------------------------------------------------------------------------
__global__ __launch_bounds__(256) void gemm1_kernel(
    const float* __restrict__ pooled_part, const float* __restrict__ cemb,
    const float* __restrict__ w1, const float* __restrict__ b1,
    float* __restrict__ hT)
{
    __shared__ float poolT[E_DIM * 33];   // [e][b], pitch 33 (33792 B)
    const int tid = threadIdx.x;

    // column e = tid: channel-embed column mean + chunk-sum reduce
    {
        float cs = 0.f;
        for (int c = 0; c < C_DIM; ++c) cs += cemb[c * E_DIM + tid];
        cs *= (1.0f / C_DIM);
        for (int b = 0; b < B_DIM; ++b) {
            float ssum = 0.f;
            for (int k = 0; k < NCH; ++k)
                ssum += pooled_part[((size_t)(k * B_DIM + b)) * E_DIM + tid];
            poolT[tid * 33 + b] = ssum * (1.0f / T_DIM) + cs;
        }
    }
    __syncthreads();

    const int wave = tid >> 5, lane = tid & 31;
    const int half = lane >> 4, idx = lane & 15;
    const int mtile = wave & 1, nt = wave >> 1;
    const int n0 = blockIdx.x * 64 + nt * 16;

    v8f acc = {};
    for (int k0 = 0; k0 < E_DIM; k0 += 4) {
        const int kk = k0 + 2 * half;
        v2f a, bf;
        a.x  = poolT[kk * 33 + mtile * 16 + idx];
        a.y  = poolT[(kk + 1) * 33 + mtile * 16 + idx];
        bf.x = w1[kk * L_DIM + n0 + idx];
        bf.y = w1[(kk + 1) * L_DIM + n0 + idx];
        acc = __builtin_amdgcn_wmma_f32_16x16x4_f32(
            false, a, false, bf, (short)0, acc, false, false);
    }

    const int col = n0 + idx;
    const float bias = b1[col];
    for (int i = 0; i < 8; ++i) {
        const int row = mtile * 16 + i + 8 * half;
        float v = acc[i] + bias;
        hT[col * B_DIM + row] = v > 0.f ? v : 0.f;   // transposed store
    }
}

// ---------------------------------------------------------------------------
// Kernel D2: GEMM2 Y = h @ W2 + b2 ([32,512]x[512,128]) with f32 WMMA.
// 1 block x 512 threads = 16 waves = exactly the 2x8 output tiles.
// A fragments read from transposed hT -> coalesced across lanes.
// ---------------------------------------------------------------------------
__global__ __launch_bounds__(512) void gemm2_kernel(
    const float* __restrict__ hT, const float* __restrict__ w2,
    const float* __restrict__ b2, float* __restrict__ Y)
{
    const int tid = threadIdx.x;
    const int wave = tid >> 5, lane = tid & 31;
    const int half = lane >> 4, idx = lane & 15;
    const int mtile = wave & 1, nt = wave >> 1;   // nt 0..7
    const int n0 = nt * 16;

    v8f acc = {};
    for (int k0 = 0; k0 < L_DIM; k0 += 4) {
        const int kk = k0 + 2 * half;
        v2f a, bf;
        a.x  = hT[kk * B_DIM + mtile * 16 + idx];
        a.y  = hT[(kk + 1) * B_DIM + mtile * 16 + idx];
        bf.x = w2[kk * C_DIM + n0 + idx];
        bf.y = w2[(kk + 1) * C_DIM + n0 + idx];
        acc = __builtin_amdgcn_wmma_f32_16x16x4_f32(
            false, a, false, bf, (short)0, acc, false, false);
    }

    const int col = n0 + idx;
    const float bias = b2[col];
    for (int i = 0; i < 8; ++i) {
        const int row = mtile * 16 + i + 8 * half;
        Y[row * C_DIM + col] = acc[i] + bias;
    }
}

// ---------------------------------------------------------------------------
// Kernel E: out[b,ty,c] = Y[b,c] + c_base[b, phase(y_times[b,ty]), c]
// 524288 elements, 2048 blocks x 256
// ---------------------------------------------------------------------------
__global__ __launch_bounds__(256) void final_kernel(
    const float* __restrict__ y_times, const float* __restrict__ Y,
    const float* __restrict__ c_base, float* __restrict__ out)
{
    const int idx = blockIdx.x * 256 + threadIdx.x;   // < B*TY*C
    const int c  = idx & (C_DIM - 1);
    const int bt = idx >> 7;
    const int ty = bt & (TY_DIM - 1);
    const int b  = bt >> 7;
    const int p  = ((int)floorf(y_times[b * TY_DIM + ty])) % PH;
    out[idx] = Y[b * C_DIM + c] + c_base[(b * PH + p) * C_DIM + c];
}

// ---------------------------------------------------------------------------
extern "C" void kernel_launch(void* const* d_in, const int* in_sizes, int n_in,
                              void* d_out, int out_size, void* d_ws, size_t ws_size,
                              hipStream_t stream) {
    const float* timesteps = (const float*)d_in[0];
    const float* X         = (const float*)d_in[1];
    const int*   M         = (const int*)  d_in[2];
    const float* y_times   = (const float*)d_in[3];
    const float* time_w    = (const float*)d_in[4];
    const float* time_b    = (const float*)d_in[5];
    const float* cemb      = (const float*)d_in[6];
    const float* dec_w1    = (const float*)d_in[7];
    const float* dec_b1    = (const float*)d_in[8];
    const float* dec_w2    = (const float*)d_in[9];
    const float* dec_b2    = (const float*)d_in[10];
    float* out = (float*)d_out;

    // workspace layout (floats)
    float* ws          = (float*)d_ws;
    float* part_sum    = ws;                                  // NCH*B*24*C = 786432
    float* part_cnt    = part_sum + (size_t)NCH * B_DIM * PH * C_DIM;
    float* c_base      = part_cnt + (size_t)NCH * B_DIM * PH * C_DIM;   // 98304
    float* pooled_part = c_base + (size_t)B_DIM * PH * C_DIM;           // NCH*B*E = 65536
    float* hT          = pooled_part + (size_t)NCH * B_DIM * E_DIM;     // L*B = 16384
    float* Yb          = hT + (size_t)L_DIM * B_DIM;                    // B*C = 4096

    cycle_accum_kernel<<<dim3(NCH, B_DIM), 256, 0, stream>>>(
        timesteps, X, M, part_sum, part_cnt);
    cbase_reduce_kernel<<<(B_DIM * PH * C_DIM) / 256, 256, 0, stream>>>(
        part_sum, part_cnt, c_base);
    time_embed_kernel<<<dim3(NCH, B_DIM), 256, 0, stream>>>(
        timesteps, time_w, time_b, pooled_part);
    gemm1_kernel<<<L_DIM / 64, 256, 0, stream>>>(
        pooled_part, cemb, dec_w1, dec_b1, hT);
    gemm2_kernel<<<1, 512, 0, stream>>>(hT, dec_w2, dec_b2, Yb);
    final_kernel<<<(B_DIM * TY_DIM * C_DIM) / 256, 256, 0, stream>>>(
        y_times, Yb, c_base, out);
}